// HeteroAttRGCNLayer_35648228556926
// MI455X (gfx1250) — compile-verified
//
#include <hip/hip_runtime.h>
#include <hip/hip_bf16.h>
#include <math.h>

typedef __attribute__((ext_vector_type(2))) float v2f;
typedef __attribute__((ext_vector_type(8))) float v8f;

#define IN_DIM 128
#define OUT_DIM 64
#define NEG_SLOPE 0.01f

// ---------------------------------------------------------------------------
// Fill helper (grid-stride)
// ---------------------------------------------------------------------------
__global__ void fill_f32(float* __restrict__ p, float v, int n) {
    int i = blockIdx.x * blockDim.x + threadIdx.x;
    int stride = gridDim.x * blockDim.x;
    for (; i < n; i += stride) p[i] = v;
}

// ---------------------------------------------------------------------------
// Wh = feat @ W + b  using V_WMMA_F32_16X16X4_F32.
// One wave -> one 16-row x 64-col output tile.
//
// W is staged in LDS pair-interleaved: Wp[(k>>1)*128 + n*2 + (k&1)], so the
// {W[k][n], W[k+1][n]} pair each lane needs is one contiguous ds_load_b64
// landing directly in an even-aligned VGPR pair (no repack movs before WMMA).
//
// f32 WMMA fragment layouts (ISA 7.12.2):
//   A 16x4 : lanes 0-15 hold M=lane,     VGPR0=K0, VGPR1=K1
//            lanes 16-31 hold M=lane-16, VGPR0=K2, VGPR1=K3
//   B 4x16 : lanes 0-15 hold N=lane (K0,K1), lanes 16-31 (K2,K3)
//   C/D    : VGPR j -> M=j (lanes 0-15) / M=j+8 (lanes 16-31), N=lane&15
// ---------------------------------------------------------------------------
__global__ void gemm_wmma_f32(const float* __restrict__ A,   // [n, 128]
                              const float* __restrict__ W,   // [128, 64]
                              const float* __restrict__ bias,// [64]
                              float* __restrict__ C,         // [n, 64]
                              int n) {
    __shared__ float Wp[(IN_DIM / 2) * (OUT_DIM * 2)]; // 32 KB, pair-interleaved
    for (int i = threadIdx.x; i < IN_DIM * OUT_DIM; i += blockDim.x) {
        const int k  = i >> 6;   // row of W (0..127)
        const int nn = i & 63;   // col of W (0..63)
        Wp[(k >> 1) * 128 + nn * 2 + (k & 1)] = W[i];
    }
    __syncthreads();

    const int lane = threadIdx.x & 31;
    const int waveInBlk = threadIdx.x >> 5;
    const int tile = blockIdx.x * (blockDim.x >> 5) + waveInBlk;
    const int m0 = tile * 16;
    if (m0 >= n) return;                    // wave-uniform exit (EXEC stays all-1)

    const int half = lane >> 4;             // 0 | 1
    const int hl   = lane & 15;
    const int mrow = min(m0 + hl, n - 1);   // clamp (n is a multiple of 16 here anyway)

    const float* Arow = A + (size_t)mrow * IN_DIM + 2 * half;
    const float* Wrow = Wp + (size_t)half * 128 + 2 * hl;

    v8f acc[4] = {};                        // 4 N-tiles of 16 -> 64 columns

    #pragma unroll 4
    for (int k0 = 0; k0 < IN_DIM; k0 += 4) {
        v2f a = *(const v2f*)(Arow + k0);
        const float* wk = Wrow + (k0 >> 1) * 128;   // (k0/2 + half) row of pairs
        #pragma unroll
        for (int nt = 0; nt < 4; ++nt) {
            v2f b = *(const v2f*)(wk + nt * 32);    // ncol = nt*16 + hl, *2 floats
            acc[nt] = __builtin_amdgcn_wmma_f32_16x16x4_f32(
                false, a, false, b, (short)0, acc[nt], false, false);
        }
    }

    #pragma unroll
    for (int nt = 0; nt < 4; ++nt) {
        const int ncol = nt * 16 + hl;
        const float bv = bias[ncol];
        #pragma unroll
        for (int j = 0; j < 8; ++j) {
            const int row = m0 + j + 8 * half;
            if (row < n) C[(size_t)row * OUT_DIM + ncol] = acc[nt][j] + bv;
        }
    }
}

// ---------------------------------------------------------------------------
// Per-node attention dots: s[n] = Wh[n,:]·a_src ; t[n] = Wh[n,:]·a_dst
// Wave per node, lane handles a float2 (coalesced 256B row read).
// ---------------------------------------------------------------------------
__global__ void node_dots(const float* __restrict__ Wh,
                          const float* __restrict__ attn, // [128] = [a_src|a_dst]
                          float* __restrict__ s_out,
                          float* __restrict__ t_out, int n) {
    const int node = (blockIdx.x * blockDim.x + threadIdx.x) >> 5;
    const int lane = threadIdx.x & 31;
    if (node >= n) return;
    v2f z  = *(const v2f*)(Wh + (size_t)node * OUT_DIM + 2 * lane);
    v2f as = *(const v2f*)(attn + 2 * lane);
    v2f ad = *(const v2f*)(attn + OUT_DIM + 2 * lane);
    float s = z.x * as.x + z.y * as.y;
    float t = z.x * ad.x + z.y * ad.y;
    #pragma unroll
    for (int off = 16; off; off >>= 1) {
        s += __shfl_xor(s, off, 32);
        t += __shfl_xor(t, off, 32);
    }
    if (lane == 0) { s_out[node] = s; t_out[node] = t; }
}

// ---------------------------------------------------------------------------
// Float atomic max via int/uint atomics (valid with -inf init).
// ---------------------------------------------------------------------------
__device__ __forceinline__ void atomicMaxFloat(float* addr, float val) {
    if (val >= 0.0f) atomicMax((int*)addr, __float_as_int(val));
    else             atomicMin((unsigned int*)addr, __float_as_uint(val));
}

__device__ __forceinline__ float leaky(float v) {
    return v > 0.0f ? v : NEG_SLOPE * v;
}

// Pass 1: segment max of leaky(e) over dst
__global__ void edge_max(const float* __restrict__ s_src,
                         const float* __restrict__ t_dst,
                         const int* __restrict__ src,
                         const int* __restrict__ dst,
                         float* __restrict__ m, int nedge) {
    int e = blockIdx.x * blockDim.x + threadIdx.x;
    if (e >= nedge) return;
    int di = dst[e];
    float v = leaky(s_src[src[e]] + t_dst[di]);
    atomicMaxFloat(&m[di], v);
}

// Pass 2: segment sum of exp(e - m[dst])
__global__ void edge_sum(const float* __restrict__ s_src,
                         const float* __restrict__ t_dst,
                         const int* __restrict__ src,
                         const int* __restrict__ dst,
                         const float* __restrict__ m,
                         float* __restrict__ ssum, int nedge) {
    int e = blockIdx.x * blockDim.x + threadIdx.x;
    if (e >= nedge) return;
    int di = dst[e];
    float v = leaky(s_src[src[e]] + t_dst[di]);
    atomicAdd(&ssum[di], expf(v - m[di]));
}

// Pass 3: h[dst] += alpha * Wh_src[src].  Wave per edge, lane moves a float2.
__global__ void edge_aggregate(const float* __restrict__ Wh_src,
                               const float* __restrict__ s_src,
                               const float* __restrict__ t_dst,
                               const int* __restrict__ src,
                               const int* __restrict__ dst,
                               const float* __restrict__ m,
                               const float* __restrict__ ssum,
                               float* __restrict__ h_out, int nedge) {
    const int e = (blockIdx.x * blockDim.x + threadIdx.x) >> 5;
    const int lane = threadIdx.x & 31;
    if (e >= nedge) return;
    const int si = src[e];
    const int di = dst[e];
    float v = leaky(s_src[si] + t_dst[di]);
    float alpha = expf(v - m[di]) / ssum[di];
    v2f z = *(const v2f*)(Wh_src + (size_t)si * OUT_DIM + 2 * lane);
    float* o = h_out + (size_t)di * OUT_DIM + 2 * lane;
    atomicAdd(o + 0, alpha * z.x);
    atomicAdd(o + 1, alpha * z.y);
}

// ---------------------------------------------------------------------------
// Host launcher
// Inputs: 0 feat_user 1 feat_item 2 W_user 3 b_user 4 W_item 5 b_item
//         6 attn_w 7 src_u2i 8 dst_u2i 9 src_i2u 10 dst_i2u
// Output: concat(h_user[nU,64], h_item[nI,64])
// ---------------------------------------------------------------------------
extern "C" void kernel_launch(void* const* d_in, const int* in_sizes, int n_in,
                              void* d_out, int out_size, void* d_ws, size_t ws_size,
                              hipStream_t stream) {
    const float* feat_user = (const float*)d_in[0];
    const float* feat_item = (const float*)d_in[1];
    const float* W_user    = (const float*)d_in[2];
    const float* b_user    = (const float*)d_in[3];
    const float* W_item    = (const float*)d_in[4];
    const float* b_item    = (const float*)d_in[5];
    const float* attn_w    = (const float*)d_in[6];
    const int*   src_u2i   = (const int*)d_in[7];
    const int*   dst_u2i   = (const int*)d_in[8];
    const int*   src_i2u   = (const int*)d_in[9];
    const int*   dst_i2u   = (const int*)d_in[10];

    const int nU = in_sizes[0] / IN_DIM;
    const int nI = in_sizes[1] / IN_DIM;
    const int nE_u2i = in_sizes[7];
    const int nE_i2u = in_sizes[9];

    float* out = (float*)d_out;
    float* h_user = out;
    float* h_item = out + (size_t)nU * OUT_DIM;

    // Workspace layout (floats)
    float* ws       = (float*)d_ws;
    float* Wh_user  = ws;
    float* Wh_item  = Wh_user + (size_t)nU * OUT_DIM;
    float* s_user   = Wh_item + (size_t)nI * OUT_DIM;
    float* t_user   = s_user + nU;
    float* s_item   = t_user + nU;
    float* t_item   = s_item + nI;
    float* m_item   = t_item + nI;
    float* sum_item = m_item + nI;
    float* m_user   = sum_item + nI;
    float* sum_user = m_user + nU;

    // ---- init (every call: deterministic, graph-replay safe) ----
    fill_f32<<<1024, 256, 0, stream>>>(out, 0.0f, (nU + nI) * OUT_DIM);
    fill_f32<<<256, 256, 0, stream>>>(m_item, -INFINITY, nI);
    fill_f32<<<256, 256, 0, stream>>>(m_user, -INFINITY, nU);
    fill_f32<<<256, 256, 0, stream>>>(sum_item, 0.0f, nI);
    fill_f32<<<256, 256, 0, stream>>>(sum_user, 0.0f, nU);

    // ---- dense projections via fp32 WMMA ----
    {
        const int wavesPerBlk = 8, thr = 32 * wavesPerBlk;
        int tilesU = (nU + 15) / 16;
        int tilesI = (nI + 15) / 16;
        gemm_wmma_f32<<<(tilesU + wavesPerBlk - 1) / wavesPerBlk, thr, 0, stream>>>(
            feat_user, W_user, b_user, Wh_user, nU);
        gemm_wmma_f32<<<(tilesI + wavesPerBlk - 1) / wavesPerBlk, thr, 0, stream>>>(
            feat_item, W_item, b_item, Wh_item, nI);
    }

    // ---- per-node attention dot products ----
    node_dots<<<(nU * 32 + 255) / 256, 256, 0, stream>>>(Wh_user, attn_w, s_user, t_user, nU);
    node_dots<<<(nI * 32 + 255) / 256, 256, 0, stream>>>(Wh_item, attn_w, s_item, t_item, nI);

    // ---- u2i: users -> items (dst = item) ----
    edge_max<<<(nE_u2i + 255) / 256, 256, 0, stream>>>(
        s_user, t_item, src_u2i, dst_u2i, m_item, nE_u2i);
    edge_sum<<<(nE_u2i + 255) / 256, 256, 0, stream>>>(
        s_user, t_item, src_u2i, dst_u2i, m_item, sum_item, nE_u2i);
    edge_aggregate<<<((size_t)nE_u2i * 32 + 255) / 256, 256, 0, stream>>>(
        Wh_user, s_user, t_item, src_u2i, dst_u2i, m_item, sum_item, h_item, nE_u2i);

    // ---- i2u: items -> users (dst = user) ----
    edge_max<<<(nE_i2u + 255) / 256, 256, 0, stream>>>(
        s_item, t_user, src_i2u, dst_i2u, m_user, nE_i2u);
    edge_sum<<<(nE_i2u + 255) / 256, 256, 0, stream>>>(
        s_item, t_user, src_i2u, dst_i2u, m_user, sum_user, nE_i2u);
    edge_aggregate<<<((size_t)nE_i2u * 32 + 255) / 256, 256, 0, stream>>>(
        Wh_item, s_item, t_user, src_i2u, dst_i2u, m_user, sum_user, h_user, nE_i2u);
}